// ForwardWarping_46531675684962
// MI455X (gfx1250) — compile-verified
//
#include <hip/hip_runtime.h>

// Forward warping with depth z-buffer (H=2160, W=3840, C=3), fp32.
// Pass 0: init zmin=+inf, cnt=0, out=0
// Pass 1: scatter-min depth bits into zmin (uint atomicMin; depth > 0)
// Pass 2: keep = d <= zmin[idx]+1.0 -> atomicAdd img into out (=num), cnt
// Pass 3: out = num / max(cnt,1)   (num==0 where cnt==0 -> stays 0)

namespace {
constexpr int H = 2160;
constexpr int W = 3840;
constexpr int N = H * W;          // 8,294,400 (divisible by 256)
constexpr float kSameRange = 1.0f;
constexpr int BLOCK = 256;
}

__global__ __launch_bounds__(BLOCK)
void fw_init(unsigned* __restrict__ zmin, float* __restrict__ cnt,
             float* __restrict__ out) {
    const int i = blockIdx.x * BLOCK + threadIdx.x;
    zmin[i] = 0x7F800000u;   // +inf bit pattern
    cnt[i]  = 0.0f;
    out[3 * i + 0] = 0.0f;   // global_store_b96
    out[3 * i + 1] = 0.0f;
    out[3 * i + 2] = 0.0f;
}

__global__ __launch_bounds__(BLOCK)
void fw_zmin(const float2* __restrict__ flow, const float* __restrict__ depth,
             unsigned* __restrict__ zmin) {
    __shared__ float sflow[BLOCK * 2];
    const int x = blockIdx.x * BLOCK + threadIdx.x;
    const int y = blockIdx.y;
    const int i = y * W + x;

    float fx, fy;
#if defined(__AMDGCN__)
    {
        // gfx1250 async copy (ASYNCcnt path): stage this wave's flow
        // (one float2 per lane) into LDS. The LDS byte offset is the low
        // 32 bits of the generic address (allocation-relative, same base
        // convention as the async instruction's dsaddr).
        const float2* g = flow + i;
        const unsigned lds_off =
            (unsigned)(size_t)(const void*)&sflow[2 * threadIdx.x];
        asm volatile("global_load_async_to_lds_b64 %0, %1, off"
                     :: "v"(lds_off), "v"(g) : "memory");
    }
    const float d = depth[i];            // overlaps with the async copy
    asm volatile("s_wait_asynccnt 0x0" ::: "memory");
    fx = sflow[2 * threadIdx.x + 0];     // ds_load_b32 (AS3 access)
    fy = sflow[2 * threadIdx.x + 1];
#else
    const float d = depth[i];
    fx = flow[i].x;
    fy = flow[i].y;
#endif

    const float tx = fminf(fmaxf((float)x + fx, 0.0f), (float)(W - 1));
    const float ty = fminf(fmaxf((float)y + fy, 0.0f), (float)(H - 1));
    const int ix = __float2int_rn(tx);   // round half-to-even == jnp.round
    const int iy = __float2int_rn(ty);
    const int idx = iy * W + ix;
    // depth > 0 -> uint ordering == float ordering
    atomicMin(zmin + idx, __float_as_uint(d));
}

__global__ __launch_bounds__(BLOCK)
void fw_scatter(const float* __restrict__ img, const float2* __restrict__ flow,
                const float* __restrict__ depth, const unsigned* __restrict__ zmin,
                float* __restrict__ num, float* __restrict__ cnt) {
    const int x = blockIdx.x * BLOCK + threadIdx.x;
    const int y = blockIdx.y;
    const int i = y * W + x;

    // global_prefetch_b8: pull the img stream while the zmin gather is in flight
    __builtin_prefetch(img + 3 * (size_t)i, 0, 0);

    const float2 f = flow[i];            // global_load_b64
    const float  d = depth[i];
    const float tx = fminf(fmaxf((float)x + f.x, 0.0f), (float)(W - 1));
    const float ty = fminf(fmaxf((float)y + f.y, 0.0f), (float)(H - 1));
    const int idx = __float2int_rn(ty) * W + __float2int_rn(tx);

    // Unconditional coalesced b96 load: most pixels survive the z-test
    // (avg ~1 contributor per cell), keep the memory pipe uniform.
    const float r = img[3 * i + 0];
    const float g = img[3 * i + 1];
    const float b = img[3 * i + 2];

    const float zm = __uint_as_float(zmin[idx]);   // random gather, L2-resident
    if (d <= zm + kSameRange) {
        // non-returning f32 atomics -> STOREcnt path, resolve in L2
        atomicAdd(num + 3 * idx + 0, r);
        atomicAdd(num + 3 * idx + 1, g);
        atomicAdd(num + 3 * idx + 2, b);
        atomicAdd(cnt + idx, 1.0f);
    }
}

__global__ __launch_bounds__(BLOCK)
void fw_norm(float* __restrict__ out, const float* __restrict__ cnt) {
    const int i = blockIdx.x * BLOCK + threadIdx.x;
    const float c  = cnt[i];
    const float cm = fmaxf(c, 1.0f);     // reference: num / max(cnt,1)
    const float r = out[3 * i + 0];
    const float g = out[3 * i + 1];
    const float b = out[3 * i + 2];
    out[3 * i + 0] = r / cm;             // IEEE div to match jnp
    out[3 * i + 1] = g / cm;
    out[3 * i + 2] = b / cm;
}

extern "C" void kernel_launch(void* const* d_in, const int* in_sizes, int n_in,
                              void* d_out, int out_size, void* d_ws, size_t ws_size,
                              hipStream_t stream) {
    (void)in_sizes; (void)n_in; (void)out_size; (void)ws_size;
    const float*  img   = (const float*)d_in[0];   // (H,W,3) f32
    const float2* flow  = (const float2*)d_in[1];  // (H,W,2) f32
    const float*  depth = (const float*)d_in[2];   // (H,W)   f32
    float* out = (float*)d_out;                    // (H,W,3) f32, doubles as num

    unsigned* zmin = (unsigned*)d_ws;                                      // N*4 B
    float*    cnt  = (float*)((char*)d_ws + (size_t)N * sizeof(unsigned)); // N*4 B

    const dim3 grid1(N / BLOCK);          // 32400 blocks
    const dim3 grid2(W / BLOCK, H);       // 15 x 2160 blocks

    fw_init   <<<grid1, BLOCK, 0, stream>>>(zmin, cnt, out);
    fw_zmin   <<<grid2, BLOCK, 0, stream>>>(flow, depth, zmin);
    fw_scatter<<<grid2, BLOCK, 0, stream>>>(img, flow, depth, zmin, out, cnt);
    fw_norm   <<<grid1, BLOCK, 0, stream>>>(out, cnt);
}